// GCN_interaction_all_86114094284911
// MI455X (gfx1250) — compile-verified
//
#include <hip/hip_runtime.h>
#include <hip/hip_bf16.h>

typedef __attribute__((ext_vector_type(2))) float v2f;
typedef __attribute__((ext_vector_type(8))) float v8f;

#define D 64

// ---------------------------------------------------------------------------
// Pass A: per-edge scatter of review features + degree + score-weight counts
// One wave (32 lanes) per edge, 2 floats per lane. Edge index forced wave-
// uniform via readfirstlane so index loads become scalar (s_load).
// review_feat is streamed once -> non-temporal loads to preserve L2 for the
// accumulator working set.
// ---------------------------------------------------------------------------
__global__ void __launch_bounds__(256)
edge_pass1_kernel(const float* __restrict__ review_feat,
                  const float* __restrict__ ci,
                  const int*   __restrict__ src,
                  const int*   __restrict__ dst,
                  const int*   __restrict__ score,
                  float* __restrict__ hre_sum,   // [N,64] accumulator
                  float* __restrict__ deg,       // [N]
                  float* __restrict__ scnt,      // [N,8] (5 used)
                  int n_edges)
{
    int e = blockIdx.x * (blockDim.x >> 5) + (threadIdx.x >> 5);
    if (e >= n_edges) return;
    e = __builtin_amdgcn_readfirstlane(e);        // wave-uniform -> scalar loads
    const int lane = threadIdx.x & 31;
    const int d = dst[e];

    const v2f* rp = (const v2f*)(review_feat + (size_t)e * D + 2 * lane);
    v2f rf = __builtin_nontemporal_load(rp);      // streamed once
    float* hp = hre_sum + (size_t)d * D + 2 * lane;
    atomicAdd(hp,     rf.x);
    atomicAdd(hp + 1, rf.y);

    if (lane == 0) {
        atomicAdd(deg + d, 1.0f);
        const int s = src[e];
        const int k = score[e] - 1;               // 1..5 -> 0..4
        atomicAdd(scnt + (size_t)d * 8 + k, ci[s]);
    }
}

// ---------------------------------------------------------------------------
// Pass B1: h_re = hre_sum / max(deg, 1)
// ---------------------------------------------------------------------------
__global__ void __launch_bounds__(256)
normalize_hre_kernel(float* __restrict__ hre, const float* __restrict__ deg,
                     int n_nodes)
{
    const size_t i = (size_t)blockIdx.x * blockDim.x + threadIdx.x;
    if (i >= (size_t)n_nodes * D) return;
    const float dg = deg[i >> 6];
    hre[i] = hre[i] / fmaxf(dg, 1.0f);
}

// ---------------------------------------------------------------------------
// Pass B2: r_fe3 = h_re @ W1^T, r_fe2 = h_re @ W2^T via V_WMMA_F32_16X16X4_F32
// Block = 128 threads = 4 waves; each wave owns one 16-col strip of a 16-row
// tile and accumulates both GEMMs (A operand reused across W1/W2).
//   A 16x4 f32 : lane<16 -> {A[M][k],A[M][k+1]}, lane>=16 -> {A[M][k+2],A[M][k+3]}
//   B 4x16 f32 : symmetric; B[k][n] = W[col0+n][k]  ->  float2 row read of W
//   C/D 16x16  : VGPR v -> row v + 8*hi, col = col0 + (lane&15)
// Out-of-range A rows are CLAMPED (not predicated) so the inner loop has no
// exec-mask churn; clamped rows only affect output rows that the store guard
// drops anyway.
// ---------------------------------------------------------------------------
__global__ void __launch_bounds__(128)
gemm_hre_wmma_kernel(const float* __restrict__ hre,   // [N,64]
                     const float* __restrict__ W1,    // [64,64] (out,in)
                     const float* __restrict__ W2,
                     float* __restrict__ rfe3,        // [N,64]
                     float* __restrict__ rfe2,
                     int n_nodes)
{
    const int wave  = threadIdx.x >> 5;           // 0..3 -> column strip
    const int lane  = threadIdx.x & 31;
    const int laneM = lane & 15;
    const int hi    = lane >> 4;                  // 0 or 1
    const int row0  = blockIdx.x * 16;
    const int col0  = wave * 16;

    int rowA = row0 + laneM;
    rowA = (rowA < n_nodes) ? rowA : (n_nodes - 1);   // clamp: branch-free
    const float* aptr  = hre + (size_t)rowA * D + 2 * hi;
    const float* b1ptr = W1  + (size_t)(col0 + laneM) * D + 2 * hi;
    const float* b2ptr = W2  + (size_t)(col0 + laneM) * D + 2 * hi;

    v8f acc1 = {};
    v8f acc2 = {};
    #pragma unroll
    for (int k = 0; k < D; k += 4) {
        v2f a  = *(const v2f*)(aptr + k);
        v2f b1 = *(const v2f*)(b1ptr + k);
        v2f b2 = *(const v2f*)(b2ptr + k);
        acc1 = __builtin_amdgcn_wmma_f32_16x16x4_f32(false, a, false, b1,
                                                     (short)0, acc1, false, false);
        acc2 = __builtin_amdgcn_wmma_f32_16x16x4_f32(false, a, false, b2,
                                                     (short)0, acc2, false, false);
    }

    #pragma unroll
    for (int v = 0; v < 8; ++v) {
        const int m = row0 + v + 8 * hi;
        if (m < n_nodes) {
            rfe3[(size_t)m * D + col0 + laneM] = acc1[v];
            rfe2[(size_t)m * D + col0 + laneM] = acc2[v];
        }
    }
}

// ---------------------------------------------------------------------------
// Pass B3: column sums of r_fe3, r_fe2, feature2, feature3 -> colsum[4][64]
// 256 threads: 4 row-groups x 64 cols; LDS reduce; 256 atomics per block.
// ---------------------------------------------------------------------------
#define CS_ROWS 512
__global__ void __launch_bounds__(256)
colsum_kernel(const float* __restrict__ rfe3, const float* __restrict__ rfe2,
              const float* __restrict__ f2,   const float* __restrict__ f3,
              float* __restrict__ colsum, int n_nodes)
{
    __shared__ float lds[4 * 256];
    const int t   = threadIdx.x;
    const int col = t & 63;
    const int g   = t >> 6;                        // 0..3
    const int r0  = blockIdx.x * CS_ROWS;
    const int rEnd = (r0 + CS_ROWS < n_nodes) ? (r0 + CS_ROWS) : n_nodes;

    float a0 = 0.f, a1 = 0.f, a2 = 0.f, a3 = 0.f;
    for (int r = r0 + g; r < rEnd; r += 4) {
        const size_t off = (size_t)r * D + col;
        a0 += rfe3[off];
        a1 += rfe2[off];
        a2 += f2[off];
        a3 += f3[off];
    }
    lds[0 * 256 + t] = a0;
    lds[1 * 256 + t] = a1;
    lds[2 * 256 + t] = a2;
    lds[3 * 256 + t] = a3;
    __syncthreads();
    if (g == 0) {
        #pragma unroll
        for (int m = 0; m < 4; ++m) {
            const float s = lds[m * 256 + col] + lds[m * 256 + 64 + col] +
                            lds[m * 256 + 128 + col] + lds[m * 256 + 192 + col];
            atomicAdd(colsum + m * D + col, s);
        }
    }
}

// ---------------------------------------------------------------------------
// Pass C: main per-edge gather/scatter (bottleneck; L2-resident atomics)
// One wave per edge, 2 floats per lane. Accumulates straight into d_out.
// ---------------------------------------------------------------------------
__global__ void __launch_bounds__(256)
edge_pass2_kernel(const float* __restrict__ ci,
                  const int*   __restrict__ src,
                  const int*   __restrict__ dst,
                  const int*   __restrict__ score,
                  const float* __restrict__ rfe3,
                  const float* __restrict__ rfe2,
                  const float* __restrict__ feature2,
                  const float* __restrict__ feature3,
                  const float* __restrict__ emb1,
                  const float* __restrict__ emb2,
                  const float* __restrict__ emb3,
                  float* __restrict__ out_rst,     // d_out slot 0
                  float* __restrict__ out_rst_re,  // d_out slot 2
                  float* __restrict__ out_rst_id,  // d_out slot 3
                  int n_edges)
{
    int e = blockIdx.x * (blockDim.x >> 5) + (threadIdx.x >> 5);
    if (e >= n_edges) return;
    e = __builtin_amdgcn_readfirstlane(e);        // wave-uniform -> scalar loads
    const int lane = threadIdx.x & 31;
    const int j = 2 * lane;

    const int s = src[e];
    const int d = dst[e];
    const int k = score[e] - 1;
    const float cis = ci[s];

    v2f x3  = *(const v2f*)(rfe3     + (size_t)s * D + j);
    v2f x2a = *(const v2f*)(feature2 + (size_t)s * D + j);
    v2f x2b = *(const v2f*)(rfe2     + (size_t)s * D + j);
    v2f xid = *(const v2f*)(feature3 + (size_t)s * D + j);
    v2f s1  = *(const v2f*)(emb1 + (size_t)k * D + j);
    v2f s2  = *(const v2f*)(emb2 + (size_t)k * D + j);
    v2f s3  = *(const v2f*)(emb3 + (size_t)k * D + j);

    float* p;
    p = out_rst_re + (size_t)d * D + j;
    atomicAdd(p,     x3.x * s1.x * cis);
    atomicAdd(p + 1, x3.y * s1.y * cis);

    p = out_rst + (size_t)d * D + j;
    atomicAdd(p,     (x2a.x + x2b.x) * s2.x * cis);
    atomicAdd(p + 1, (x2a.y + x2b.y) * s2.y * cis);

    p = out_rst_id + (size_t)d * D + j;
    atomicAdd(p,     xid.x * s3.x * cis);
    atomicAdd(p + 1, xid.y * s3.y * cis);
}

// ---------------------------------------------------------------------------
// Pass D: finalize. One thread per (node, j).
//   slots 0/2/3: *= ci[n] in place
//   slot 1: ci*(f2_mean+r2_mean) ⊙ (scnt@emb2)
//   slot 4: ci*r3_mean ⊙ (scnt@emb1)
//   slot 5: ci*f3_mean ⊙ (scnt@emb3)
// Freeze slots are written once and never re-read -> non-temporal stores.
// ---------------------------------------------------------------------------
__global__ void __launch_bounds__(256)
finalize_kernel(const float* __restrict__ ci,
                const float* __restrict__ scnt,     // [N,8]
                const float* __restrict__ emb1,
                const float* __restrict__ emb2,
                const float* __restrict__ emb3,
                const float* __restrict__ colsum,   // [4][64] sums
                float* __restrict__ out, int n_nodes)
{
    const size_t i = (size_t)blockIdx.x * blockDim.x + threadIdx.x;
    const size_t ND = (size_t)n_nodes * D;
    if (i >= ND) return;
    const int n = (int)(i >> 6);
    const int j = (int)(i & 63);

    const float c    = ci[n];
    const float invN = 1.0f / (float)n_nodes;
    const float r3m  = colsum[0 * D + j] * invN;
    const float r2m  = colsum[1 * D + j] * invN;
    const float f2m  = colsum[2 * D + j] * invN;
    const float f3m  = colsum[3 * D + j] * invN;

    const float w0 = scnt[(size_t)n * 8 + 0];
    const float w1 = scnt[(size_t)n * 8 + 1];
    const float w2 = scnt[(size_t)n * 8 + 2];
    const float w3 = scnt[(size_t)n * 8 + 3];
    const float w4 = scnt[(size_t)n * 8 + 4];

    const float A1 = w0*emb1[j] + w1*emb1[D+j] + w2*emb1[2*D+j] + w3*emb1[3*D+j] + w4*emb1[4*D+j];
    const float A2 = w0*emb2[j] + w1*emb2[D+j] + w2*emb2[2*D+j] + w3*emb2[3*D+j] + w4*emb2[4*D+j];
    const float A3 = w0*emb3[j] + w1*emb3[D+j] + w2*emb3[2*D+j] + w3*emb3[3*D+j] + w4*emb3[4*D+j];

    out[0 * ND + i] *= c;                                            // rst
    __builtin_nontemporal_store(c * (f2m + r2m) * A2, &out[1 * ND + i]); // rst_freeze
    out[2 * ND + i] *= c;                                            // rst_re
    out[3 * ND + i] *= c;                                            // rst_id
    __builtin_nontemporal_store(c * r3m * A1, &out[4 * ND + i]);     // rst_re_freeze
    __builtin_nontemporal_store(c * f3m * A3, &out[5 * ND + i]);     // rst_id_freeze
}

// ---------------------------------------------------------------------------
extern "C" void kernel_launch(void* const* d_in, const int* in_sizes, int n_in,
                              void* d_out, int out_size, void* d_ws, size_t ws_size,
                              hipStream_t stream) {
    const float* review_feat = (const float*)d_in[0];   // [E,64]
    const float* ci          = (const float*)d_in[1];   // [N,1]
    const float* W1          = (const float*)d_in[2];   // [64,64]
    const float* W2          = (const float*)d_in[3];   // [64,64]
    const float* emb1        = (const float*)d_in[4];   // [5,64]
    const float* emb2        = (const float*)d_in[5];
    const float* emb3        = (const float*)d_in[6];
    const float* feature2    = (const float*)d_in[7];   // [N,64]
    const float* feature3    = (const float*)d_in[8];
    const int*   src         = (const int*)d_in[9];     // [E]
    const int*   dst         = (const int*)d_in[10];
    const int*   score       = (const int*)d_in[11];

    const int n_nodes = in_sizes[1];            // ci has N elements
    const int n_edges = in_sizes[9];
    const size_t ND   = (size_t)n_nodes * D;

    // Workspace layout (floats): hre | rfe3 | rfe2 | deg | scnt | colsum
    float* ws     = (float*)d_ws;
    float* hre    = ws;                 // ND   (accumulator -> normalized in place)
    float* rfe3   = ws + ND;            // ND
    float* rfe2   = ws + 2 * ND;        // ND
    float* deg    = ws + 3 * ND;        // N
    float* scnt   = deg + n_nodes;      // 8*N
    float* colsum = scnt + 8 * (size_t)n_nodes;  // 256

    float* out = (float*)d_out;

    // Zero only what is accumulated into (slots 1/4/5 are fully overwritten):
    hipMemsetAsync(out, 0, ND * sizeof(float), stream);              // slot 0
    hipMemsetAsync(out + 2 * ND, 0, 2 * ND * sizeof(float), stream); // slots 2,3
    hipMemsetAsync(hre, 0, ND * sizeof(float), stream);
    hipMemsetAsync(deg, 0, ((size_t)n_nodes * 9 + 256) * sizeof(float), stream);

    const int waves_per_block = 8;       // 256 threads
    const int edge_blocks = (n_edges + waves_per_block - 1) / waves_per_block;

    edge_pass1_kernel<<<edge_blocks, 256, 0, stream>>>(
        review_feat, ci, src, dst, score, hre, deg, scnt, n_edges);

    const int nd_blocks = (int)((ND + 255) / 256);
    normalize_hre_kernel<<<nd_blocks, 256, 0, stream>>>(hre, deg, n_nodes);

    const int gemm_blocks = (n_nodes + 15) / 16;
    gemm_hre_wmma_kernel<<<gemm_blocks, 128, 0, stream>>>(
        hre, W1, W2, rfe3, rfe2, n_nodes);

    const int cs_blocks = (n_nodes + CS_ROWS - 1) / CS_ROWS;
    colsum_kernel<<<cs_blocks, 256, 0, stream>>>(
        rfe3, rfe2, feature2, feature3, colsum, n_nodes);

    edge_pass2_kernel<<<edge_blocks, 256, 0, stream>>>(
        ci, src, dst, score, rfe3, rfe2, feature2, feature3,
        emb1, emb2, emb3,
        out + 0 * ND,   // rst
        out + 2 * ND,   // rst_re
        out + 3 * ND,   // rst_id
        n_edges);

    finalize_kernel<<<nd_blocks, 256, 0, stream>>>(
        ci, scnt, emb1, emb2, emb3, colsum, out, n_nodes);
}